// SAGE_2585570312619
// MI455X (gfx1250) — compile-verified
//
#include <hip/hip_runtime.h>

#define N_NODES_C 100000
#define FEAT_C 64
#define HID_C 128
#define N_EDGES_C 1200000

typedef __attribute__((ext_vector_type(2))) float v2f;
typedef __attribute__((ext_vector_type(8))) float v8f;

// ---------------------------------------------------------------------------
// Zero a float region (grid-stride). Needed every launch: atomics accumulate.
// ---------------------------------------------------------------------------
__global__ void sage_zero_f32(float* __restrict__ p, long n) {
    long i = (long)blockIdx.x * blockDim.x + threadIdx.x;
    long stride = (long)gridDim.x * blockDim.x;
    for (; i < n; i += stride) p[i] = 0.0f;
}

// ---------------------------------------------------------------------------
// deg[dst[e]] += 1.0f
// ---------------------------------------------------------------------------
__global__ void sage_degree(const int* __restrict__ dst, float* __restrict__ deg, int E) {
    int e = blockIdx.x * blockDim.x + threadIdx.x;
    if (e < E) atomicAdd(&deg[dst[e]], 1.0f);
}

// ---------------------------------------------------------------------------
// agg[dst[e]*D + f] += x[src[e]*D + f]; thread = (edge, feature) pair.
// Consecutive D threads share an edge -> coalesced row reads; atomics resolve
// in L2 (entire ~165MB working set fits the 192MB L2 on MI455X).
// ---------------------------------------------------------------------------
template <int D>
__global__ void sage_scatter(const float* __restrict__ x, const int* __restrict__ src,
                             const int* __restrict__ dst, float* __restrict__ agg, int E) {
    long idx = (long)blockIdx.x * blockDim.x + threadIdx.x;
    long total = (long)E * D;
    if (idx >= total) return;
    int e = (int)(idx / D);
    int f = (int)(idx % D);
    int s = src[e];
    int d = dst[e];
    atomicAdd(&agg[(long)d * D + f], x[(long)s * D + f]);
}

// ---------------------------------------------------------------------------
// out[16 rows x NT*16 cols] = relu?( (agg/deg) @ Wl^T + bl + xr @ Wr^T )
// 8 waves per block; each wave owns one 16-row tile across ALL column tiles.
// Weights are staged once per block into padded LDS (KP = K+2 -> lane stride
// of 2 banks, conflict-free for the 16-lane half-wave, 8B aligned for b64).
// Full-precision V_WMMA_F32_16X16X4_F32.
//
// A 16x4 f32 layout (ISA 7.12.2): lane L holds row M=L%16;
//   lanes 0-15: vgpr0=K+0, vgpr1=K+1;  lanes 16-31: vgpr0=K+2, vgpr1=K+3.
// B 4x16 mirrors (N=L%16 striped across lanes, K split by lane half).
// C/D 16x16 f32: vgpr v, lane L -> M = v + 8*(L/16), N = L%16.
// ---------------------------------------------------------------------------
template <int K, int NT, bool RELU>
__global__ void __launch_bounds__(256)
sage_gemm(const float* __restrict__ agg, const float* __restrict__ deg,
          const float* __restrict__ xr, const float* __restrict__ Wl,
          const float* __restrict__ bl, const float* __restrict__ Wr,
          float* __restrict__ out, int n_nodes) {
    constexpr int NC = NT * 16;  // output columns
    constexpr int KP = K + 2;    // padded LDS row stride (floats)

    __shared__ float sW[2 * NC * KP];
    float* sWl = sW;
    float* sWr = sW + NC * KP;

    // Cooperative, coalesced staging of both weight matrices into LDS.
    for (int i = threadIdx.x; i < NC * K; i += blockDim.x) {
        int r = i / K;
        int kk = i - r * K;
        sWl[r * KP + kk] = Wl[i];
        sWr[r * KP + kk] = Wr[i];
    }
    __syncthreads();

    const int wave = threadIdx.x >> 5;
    const int lane = threadIdx.x & 31;
    const int tile_m = blockIdx.x * (blockDim.x >> 5) + wave;
    if (tile_m * 16 >= n_nodes) return;           // wave-uniform: EXEC stays all-1s

    const int mrow = tile_m * 16 + (lane & 15);   // M row owned by this lane (A side)
    const int khalf = (lane >> 4) * 2;            // 0 for lanes 0-15, 2 for lanes 16-31
    const int col_lane = lane & 15;               // N position within a col tile

    const float dv = deg[mrow];
    const float invd = 1.0f / fmaxf(dv, 1.0f);

    v8f acc[NT] = {};

    const float* arow = agg + (long)mrow * K;
    const float* xrow = xr + (long)mrow * K;

    for (int k = 0; k < K; k += 4) {
        float2 ta = *(const float2*)(arow + k + khalf);
        float2 tx = *(const float2*)(xrow + k + khalf);
        v2f a_agg; a_agg[0] = ta.x * invd; a_agg[1] = ta.y * invd;
        v2f a_x;   a_x[0] = tx.x;          a_x[1] = tx.y;
#pragma unroll
        for (int c = 0; c < NT; c++) {
            const int col = c * 16 + col_lane;
            const v2f bL = *(const v2f*)(sWl + col * KP + k + khalf);
            const v2f bR = *(const v2f*)(sWr + col * KP + k + khalf);
            acc[c] = __builtin_amdgcn_wmma_f32_16x16x4_f32(
                false, a_agg, false, bL, (short)0, acc[c], false, false);
            acc[c] = __builtin_amdgcn_wmma_f32_16x16x4_f32(
                false, a_x, false, bR, (short)0, acc[c], false, false);
        }
    }

    const int mbase = tile_m * 16 + (lane >> 4) * 8;
#pragma unroll
    for (int c = 0; c < NT; c++) {
        const int col = c * 16 + col_lane;
        const float bias = bl[col];
#pragma unroll
        for (int v = 0; v < 8; v++) {
            float val = acc[c][v] + bias;
            if (RELU) val = fmaxf(val, 0.0f);
            out[(long)(mbase + v) * NC + col] = val;
        }
    }
}

// ---------------------------------------------------------------------------
extern "C" void kernel_launch(void* const* d_in, const int* in_sizes, int n_in,
                              void* d_out, int out_size, void* d_ws, size_t ws_size,
                              hipStream_t stream) {
    (void)in_sizes; (void)n_in; (void)out_size; (void)ws_size;

    const float* x   = (const float*)d_in[0];
    const int*   ei  = (const int*)d_in[1];   // [2, E]: row0 = src, row1 = dst
    const float* W1l = (const float*)d_in[2]; // [HID, FEAT]
    const float* b1l = (const float*)d_in[3]; // [HID]
    const float* W1r = (const float*)d_in[4]; // [HID, FEAT]
    const float* W2l = (const float*)d_in[5]; // [FEAT, HID]
    const float* b2l = (const float*)d_in[6]; // [FEAT]
    const float* W2r = (const float*)d_in[7]; // [FEAT, HID]
    float* out = (float*)d_out;               // [N, FEAT]

    const int* src = ei;
    const int* dst = ei + N_EDGES_C;

    // Workspace layout (floats): deg[N] | agg1[N*FEAT] | h[N*HID] | agg2[N*HID]
    float* deg  = (float*)d_ws;
    float* agg1 = deg + N_NODES_C;
    float* h    = agg1 + (long)N_NODES_C * FEAT_C;
    float* agg2 = h + (long)N_NODES_C * HID_C;

    // 1) zero accumulators (deg+agg1 contiguous; agg2 separate)
    {
        long n1 = (long)N_NODES_C * (1 + FEAT_C);
        long n2 = (long)N_NODES_C * HID_C;
        sage_zero_f32<<<2048, 256, 0, stream>>>(deg, n1);
        sage_zero_f32<<<2048, 256, 0, stream>>>(agg2, n2);
    }

    // 2) degrees
    sage_degree<<<(N_EDGES_C + 255) / 256, 256, 0, stream>>>(dst, deg, N_EDGES_C);

    // 3) layer-1 scatter: agg1 += x[src]
    {
        long total = (long)N_EDGES_C * FEAT_C;
        int blocks = (int)((total + 255) / 256);
        sage_scatter<FEAT_C><<<blocks, 256, 0, stream>>>(x, src, dst, agg1, N_EDGES_C);
    }

    // 4) layer-1 GEMM: h = relu((agg1/deg) @ W1l^T + b1l + x @ W1r^T)
    {
        int tiles = (N_NODES_C + 15) / 16;       // 6250
        int blocks = (tiles + 7) / 8;            // 8 waves per block
        sage_gemm<FEAT_C, HID_C / 16, true><<<blocks, 256, 0, stream>>>(
            agg1, deg, x, W1l, b1l, W1r, h, N_NODES_C);
    }

    // 5) layer-2 scatter: agg2 += h[src]
    {
        long total = (long)N_EDGES_C * HID_C;
        int blocks = (int)((total + 255) / 256);
        sage_scatter<HID_C><<<blocks, 256, 0, stream>>>(h, src, dst, agg2, N_EDGES_C);
    }

    // 6) layer-2 GEMM: out = (agg2/deg) @ W2l^T + b2l + h @ W2r^T
    {
        int tiles = (N_NODES_C + 15) / 16;
        int blocks = (tiles + 7) / 8;
        sage_gemm<HID_C, FEAT_C / 16, false><<<blocks, 256, 0, stream>>>(
            agg2, deg, h, W2l, b2l, W2r, out, N_NODES_C);
    }
}